// Net_57664230916956
// MI455X (gfx1250) — compile-verified
//
#include <hip/hip_runtime.h>

// SNN forward scan: 80 steps, batch 65536, 20->10->2, LIF (beta=0.5, thr=1, subtract reset).
// One wave32 owns 16 batch rows; mem1 kept in WMMA C-layout across the whole time loop.
// x loads are software-pipelined one step ahead + global_prefetch 6 steps ahead.

typedef float v2f __attribute__((ext_vector_type(2)));
typedef float v8f __attribute__((ext_vector_type(8)));

#define NUM_STEPS 80
#define BATCH     65536
#define NUM_IN    20
#define NUM_HID   10
#define NUM_OUT   2
#define PF_DIST   6

__global__ __launch_bounds__(256) void snn_fwd_kernel(
    const float* __restrict__ x,   // [80, 65536, 20]
    const float* __restrict__ W1,  // [10, 20]
    const float* __restrict__ b1,  // [10]
    const float* __restrict__ W2,  // [2, 10]
    const float* __restrict__ b2,  // [2]
    float* __restrict__ out)       // [80,65536,2] spk2 then [80,65536,2] mem2
{
    __shared__ float lut[1024 * 2];  // lut[mask*2 + j] = sum_n bit_n(mask) * W2[j][n]

    const int tid  = threadIdx.x;
    const int lane = tid & 31;
    const int half = lane >> 4;   // 0: lanes 0-15, 1: lanes 16-31 (also output index j)
    const int m    = lane & 15;   // batch row within tile (layer2/store); also N col in C layout

    // ---- Build layer-2 LUT once per block (W2 loads are uniform -> scalar cached) ----
    for (int i = tid; i < 1024; i += 256) {
        float s0 = 0.f, s1 = 0.f;
        #pragma unroll
        for (int n = 0; n < NUM_HID; ++n) {
            float bit = (float)((i >> n) & 1);
            s0 += bit * W2[n];
            s1 += bit * W2[NUM_HID + n];
        }
        lut[i * 2 + 0] = s0;
        lut[i * 2 + 1] = s1;
    }
    __syncthreads();

    const int wave = blockIdx.x * 8 + (tid >> 5);
    const int b0   = wave * 16;   // first batch row of this wave's tile

    // ---- B-matrix slices of W1^T (4x16 each, K=20 -> 5 slices), resident for whole loop.
    // B layout (k,n): lane = n + 16*(k>=2 within slice), vgpr = k%2.
    v2f bw[5];
    #pragma unroll
    for (int s = 0; s < 5; ++s) {
        const int k0 = 4 * s + 2 * half;
        bw[s].x = (m < NUM_HID) ? W1[m * NUM_IN + k0]     : 0.f;
        bw[s].y = (m < NUM_HID) ? W1[m * NUM_IN + k0 + 1] : 0.f;
    }
    const float b1l = (m < NUM_HID) ? b1[m] : 0.f;  // per-lane bias for C column N=m
    const float b2l = b2[half];

    // Row-select predicates for mask extraction (fixed per lane)
    const bool rb0 = (m & 1) != 0;
    const bool rb1 = (m & 2) != 0;
    const bool rb2 = (m & 4) != 0;
    const bool rhi = (m & 8) != 0;

    // ---- Membrane state ----
    float m1[8];                      // mem1 in C layout: (row = r + 8*half, hid = m)
    #pragma unroll
    for (int r = 0; r < 8; ++r) m1[r] = 0.f;
    float mem2 = 0.f;                 // (row = m, out = half)

    float* __restrict__ outS = out;
    float* __restrict__ outM = out + (size_t)NUM_STEPS * BATCH * NUM_OUT;

    // A-slice loader: lane holds row m, K pair selected by lane half.
    auto loadx = [&](int t, v2f* a) {
        const float* xt = x + ((size_t)t * BATCH + (size_t)(b0 + m)) * NUM_IN + 2 * half;
        #pragma unroll
        for (int s = 0; s < 5; ++s)
            a[s] = __builtin_nontemporal_load((const v2f*)(xt + 4 * s));
    };

    // Prefetch a future step's 1280B tile region: 32 lanes at 40B stride cover every line.
    auto pref = [&](int t) {
        int tp = (t < NUM_STEPS) ? t : (NUM_STEPS - 1);
        const float* p = x + ((size_t)tp * BATCH + (size_t)b0) * NUM_IN + lane * 10;
        __builtin_prefetch(p, 0, 3);
    };

    // One LIF step given preloaded A slices.
    auto step = [&](int t, const v2f* a) {
        // C0 = beta*mem1 + b1 - reset1  (reset uses OLD mem1)
        v8f c;
        #pragma unroll
        for (int r = 0; r < 8; ++r) {
            float reset = (m1[r] > 1.0f) ? 1.0f : 0.0f;
            c[r] = 0.5f * m1[r] + b1l - reset;
        }

        // mem1_new = C0 + x_t @ W1^T   (5 chained f32 WMMAs, K=4 each)
        #pragma unroll
        for (int s = 0; s < 5; ++s)
            c = __builtin_amdgcn_wmma_f32_16x16x4_f32(
                    false, a[s], false, bw[s], (short)0, c, false, false);

        // spikes -> per-VGPR wave-uniform bitmasks
        unsigned int mk[8];
        #pragma unroll
        for (int r = 0; r < 8; ++r) {
            m1[r] = c[r];
            mk[r] = __builtin_amdgcn_ballot_w32(c[r] > 1.0f);
        }

        // Extract this lane's row mask: bits n=0..9 of (row<8 ? mk[row] : mk[row-8]>>16)
        unsigned int s0 = rb0 ? mk[1] : mk[0];
        unsigned int s1 = rb0 ? mk[3] : mk[2];
        unsigned int s2 = rb0 ? mk[5] : mk[4];
        unsigned int s3 = rb0 ? mk[7] : mk[6];
        unsigned int t0 = rb1 ? s1 : s0;
        unsigned int t1 = rb1 ? s3 : s2;
        unsigned int mysel = rb2 ? t1 : t0;
        unsigned int bits = (rhi ? (mysel >> 16) : mysel) & 0x3FFu;

        // Layer 2 via LUT, then LIF on mem2
        float cur2   = b2l + lut[bits * 2 + half];
        float reset2 = (mem2 > 1.0f) ? 1.0f : 0.0f;
        mem2 = 0.5f * mem2 + cur2 - reset2;
        float spk2 = (mem2 > 1.0f) ? 1.0f : 0.0f;

        // Store spk2 / mem2: wave covers a contiguous 32-dword region
        size_t oidx = ((size_t)t * BATCH + (size_t)(b0 + m)) * NUM_OUT + half;
        __builtin_nontemporal_store(spk2, outS + oidx);
        __builtin_nontemporal_store(mem2, outM + oidx);
    };

    // ---- Main scan, software-pipelined x loads (ping-pong, unroll x2, no copies) ----
    v2f aA[5], aB[5];
    loadx(0, aA);
    #pragma unroll 1
    for (int t = 0; t < NUM_STEPS; t += 2) {
        loadx(t + 1, aB);                                  // prefetch t+1 into regs
        pref(t + PF_DIST);
        step(t, aA);
        loadx((t + 2 < NUM_STEPS) ? (t + 2) : (t + 1), aB[0].x == aB[0].x ? aA : aA); // t+2 -> aA
        pref(t + 1 + PF_DIST);
        step(t + 1, aB);
    }
}

extern "C" void kernel_launch(void* const* d_in, const int* in_sizes, int n_in,
                              void* d_out, int out_size, void* d_ws, size_t ws_size,
                              hipStream_t stream) {
    const float* x  = (const float*)d_in[0];
    const float* W1 = (const float*)d_in[1];
    const float* b1 = (const float*)d_in[2];
    const float* W2 = (const float*)d_in[3];
    const float* b2 = (const float*)d_in[4];
    float* out = (float*)d_out;

    // 16 batch rows per wave, 8 waves per block -> 128 rows per block
    dim3 grid(BATCH / (16 * 8));   // 512 blocks
    dim3 block(256);
    snn_fwd_kernel<<<grid, block, 0, stream>>>(x, W1, b1, W2, b2, out);
}